// QuantumConvolutionalLayer_5652176962204
// MI455X (gfx1250) — compile-verified
//
#include <hip/hip_runtime.h>

typedef float v2f __attribute__((ext_vector_type(2)));
typedef float v8f __attribute__((ext_vector_type(8)));

#define HWDIM 16129   // 127*127
#define WPB   6       // waves per block
#define LDSS  17      // padded row stride for phi tile

// ---------------------------------------------------------------------------
// Setup kernel: build A = Re(U^H Z0 U), U = fixed 2-layer circuit from weights.
// Threads 0..15 each evolve one basis column in registers; then 256 threads
// form the 16x16 symmetric matrix A into d_ws.
// ---------------------------------------------------------------------------
__global__ __launch_bounds__(256) void build_A_kernel(const float* __restrict__ w,
                                                      float* __restrict__ Amat) {
    __shared__ float2 Ush[16][16];   // Ush[m][col]
    const int t = threadIdx.x;

    if (t < 16) {
        float2 u[16];
#pragma unroll
        for (int mm = 0; mm < 16; ++mm) u[mm] = make_float2(mm == t ? 1.f : 0.f, 0.f);

#pragma unroll
        for (int layer = 0; layer < 2; ++layer) {
#pragma unroll
            for (int q = 0; q < 4; ++q) {
                const float thx = w[(layer * 4 + q) * 3 + 0];
                const float thy = w[(layer * 4 + q) * 3 + 1];
                const float thz = w[(layer * 4 + q) * 3 + 2];
                const int msk = 1 << (3 - q);
                // RX: [[c, -i s],[-i s, c]]
                {
                    float c = __cosf(0.5f * thx), s = __sinf(0.5f * thx);
#pragma unroll
                    for (int mm = 0; mm < 16; ++mm) if (!(mm & msk)) {
                        float2 a = u[mm], b = u[mm | msk];
                        u[mm]       = make_float2(c * a.x + s * b.y, c * a.y - s * b.x);
                        u[mm | msk] = make_float2(s * a.y + c * b.x, c * b.y - s * a.x);
                    }
                }
                // RY: [[c, -s],[s, c]]
                {
                    float c = __cosf(0.5f * thy), s = __sinf(0.5f * thy);
#pragma unroll
                    for (int mm = 0; mm < 16; ++mm) if (!(mm & msk)) {
                        float2 a = u[mm], b = u[mm | msk];
                        u[mm]       = make_float2(c * a.x - s * b.x, c * a.y - s * b.y);
                        u[mm | msk] = make_float2(s * a.x + c * b.x, s * a.y + c * b.y);
                    }
                }
                // RZ: diag(e^{-i th/2}, e^{+i th/2})
                {
                    float c = __cosf(0.5f * thz), s = __sinf(0.5f * thz);
#pragma unroll
                    for (int mm = 0; mm < 16; ++mm) if (!(mm & msk)) {
                        float2 a = u[mm], b = u[mm | msk];
                        u[mm]       = make_float2(a.x * c + a.y * s, a.y * c - a.x * s);
                        u[mm | msk] = make_float2(b.x * c - b.y * s, b.y * c + b.x * s);
                    }
                }
            }
            // CNOT(0,1), CNOT(2,3), CNOT(0,2), CNOT(1,3)
            const int cw[4] = {0, 2, 0, 1};
            const int tw[4] = {1, 3, 2, 3};
#pragma unroll
            for (int e = 0; e < 4; ++e) {
                const int mc = 1 << (3 - cw[e]);
                const int mt = 1 << (3 - tw[e]);
#pragma unroll
                for (int mm = 0; mm < 16; ++mm) if ((mm & mc) && !(mm & mt)) {
                    float2 tmp = u[mm]; u[mm] = u[mm | mt]; u[mm | mt] = tmp;
                }
            }
        }
#pragma unroll
        for (int mm = 0; mm < 16; ++mm) Ush[mm][t] = u[mm];
    }
    __syncthreads();

    const int jj = t >> 4, kk = t & 15;
    float acc = 0.f;
#pragma unroll
    for (int mm = 0; mm < 16; ++mm) {
        const float zz = (mm & 8) ? -1.f : 1.f;
        acc += zz * (Ush[mm][jj].x * Ush[mm][kk].x + Ush[mm][jj].y * Ush[mm][kk].y);
    }
    Amat[jj * 16 + kk] = acc;
}

// ---------------------------------------------------------------------------
// Main kernel: one 16-patch tile per wave. ev = rowsum( Phi  .*  (Phi x A) )
// GEMM done with 4x V_WMMA_F32_16X16X4_F32 (full f32 precision).
// ---------------------------------------------------------------------------
__global__ __launch_bounds__(192) void qconv_wmma_kernel(const float* __restrict__ x,
                                                         const float* __restrict__ Amat,
                                                         float* __restrict__ out) {
    __shared__ float phiLds[WPB][16 * LDSS];
    __shared__ float evLds[WPB][16];

    const int wave = threadIdx.x >> 5;
    const int lane = threadIdx.x & 31;
    const int half = lane >> 4;      // which half-wave
    const int m    = lane & 15;      // patch row within tile (mirrored across halves)

    const unsigned tile = blockIdx.x * WPB + wave;
    const unsigned base = tile * 16u;
    const unsigned p    = base + (unsigned)m;

    // decode patch -> (channel-image, i, j); x is [96,128,128]
    const unsigned cimg = p / HWDIM;
    const unsigned rem  = p - cimg * HWDIM;
    const unsigned i    = rem / 127u;
    const unsigned j    = rem - i * 127u;
    const float* px = x + ((size_t)cimg << 14) + ((size_t)i << 7) + (size_t)j;

    const float a0 = px[0],   a1 = px[1];
    const float a2 = px[128], a3 = px[129];

    const float c0 = __cosf(0.5f * a0), s0 = __sinf(0.5f * a0);
    const float c1 = __cosf(0.5f * a1), s1 = __sinf(0.5f * a1);
    const float c2 = __cosf(0.5f * a2), s2 = __sinf(0.5f * a2);
    const float c3 = __cosf(0.5f * a3), s3 = __sinf(0.5f * a3);

    // phi[k] = u0[b0]*u1[b1]*u2[b2]*u3[b3], k = b0*8+b1*4+b2*2+b3
    const float p01[4] = {c0 * c1, c0 * s1, s0 * c1, s0 * s1};
    const float p23[4] = {c2 * c3, c2 * s3, s2 * c3, s2 * s3};
    float phi[16];
#pragma unroll
    for (int k = 0; k < 16; ++k) phi[k] = p01[k >> 2] * p23[k & 3];

    // stash Phi tile for the post-GEMM row-dot (one copy per tile)
    if (lane < 16) {
#pragma unroll
        for (int k = 0; k < 16; ++k) phiLds[wave][m * LDSS + k] = phi[k];
    }

    // Y = Phi (16x16, M x K) * A (16x16, K x N) via four 16x16x4 f32 WMMAs.
    // f32 operand layout: half-wave splits the K-chunk in half, VGPR index
    // advances K within the half; M (for A-op) / N (for B-op) = lane&15.
    v8f acc = {0.f, 0.f, 0.f, 0.f, 0.f, 0.f, 0.f, 0.f};
#pragma unroll
    for (int kc = 0; kc < 4; ++kc) {
        const int kb = kc * 4 + 2 * half;
        v2f afrag = {phi[kb], phi[kb + 1]};
        v2f bfrag = {Amat[kb * 16 + m], Amat[(kb + 1) * 16 + m]};
        acc = __builtin_amdgcn_wmma_f32_16x16x4_f32(
            /*neg_a=*/false, afrag, /*neg_b=*/false, bfrag,
            /*c_mod=*/(short)0, acc, /*reuse_a=*/false, /*reuse_b=*/false);
    }

    __syncthreads();

    // ev[mrow] = sum_n Phi[mrow][n] * Y[mrow][n]
    // D layout: VGPR r holds row (r + 8*half), lane within half = column n.
    float ev[8];
#pragma unroll
    for (int r = 0; r < 8; ++r) {
        float tt = phiLds[wave][(r + 8 * half) * LDSS + m] * acc[r];
        tt += __shfl_xor(tt, 1);
        tt += __shfl_xor(tt, 2);
        tt += __shfl_xor(tt, 4);
        tt += __shfl_xor(tt, 8);
        ev[r] = tt;
    }
    if (m == 0) {          // lane 0 -> rows 0..7, lane 16 -> rows 8..15
#pragma unroll
        for (int r = 0; r < 8; ++r) evLds[wave][half * 8 + r] = ev[r];
    }
    __syncthreads();

    if (lane < 16) out[base + lane] = evLds[wave][lane];
}

// ---------------------------------------------------------------------------
extern "C" void kernel_launch(void* const* d_in, const int* in_sizes, int n_in,
                              void* d_out, int out_size, void* d_ws, size_t ws_size,
                              hipStream_t stream) {
    const float* x = (const float*)d_in[0];   // [32,3,128,128] f32
    const float* w = (const float*)d_in[1];   // [2,4,3] f32
    float* out  = (float*)d_out;              // [32,3,127,127] f32
    float* Amat = (float*)d_ws;               // 256 floats scratch

    build_A_kernel<<<1, 256, 0, stream>>>(w, Amat);

    const int ntiles  = out_size / 16;        // 96774, divisible by WPB=6
    const int nblocks = ntiles / WPB;         // 16129
    qconv_wmma_kernel<<<nblocks, 192, 0, stream>>>(x, Amat, out);
}